// Prototype_46445776339034
// MI455X (gfx1250) — compile-verified
//
#include <hip/hip_runtime.h>
#include <hip/hip_bf16.h>

typedef __attribute__((ext_vector_type(2))) float v2f;
typedef __attribute__((ext_vector_type(8))) float v8f;

#define B_SZ 256
#define L_SZ 1024
#define D_SZ 768
#define H_SZ 768
#define M_ROWS 2048   // 1024 pos rows + 1024 anti rows

// ---------------------------------------------------------------------------
// Kernel 1: masked segment reduction over B, fused mean/prototype-average.
// One block per l (1024 blocks, 256 threads, 3 columns each).
// Writes Amat[2048 x 768]: rows [0,1024) = avg_pos, rows [1024,2048) = avg_anti.
// Memory-bound: one coalesced pass over x (805 MB) -> ~35us at 23.3 TB/s.
// ---------------------------------------------------------------------------
__global__ __launch_bounds__(256) void reduce_kernel(
    const float* __restrict__ x,        // [B, L, D]
    const int*   __restrict__ mask,     // [B, L]
    const float* __restrict__ G,        // [L+1, D]
    float* __restrict__ Amat,           // [2048, 768]
    float* __restrict__ posCnt)         // [1024]
{
    const int l   = blockIdx.x;
    const int tid = threadIdx.x;

    float tot0 = 0.f, tot1 = 0.f, tot2 = 0.f;
    float pos0 = 0.f, pos1 = 0.f, pos2 = 0.f;
    int cnt = 0;

    const float* xp = x + (size_t)l * D_SZ;
    const size_t bstride = (size_t)L_SZ * D_SZ;

    for (int b = 0; b < B_SZ; ++b) {
        const int mv = mask[b * L_SZ + l];
        const float v0 = xp[tid];
        const float v1 = xp[tid + 256];
        const float v2 = xp[tid + 512];
        // prefetch next b-row of this l (speculative; dropped if OOB)
        __builtin_prefetch(xp + bstride + tid, 0, 0);
        tot0 += v0; tot1 += v1; tot2 += v2;
        if (mv) { pos0 += v0; pos1 += v1; pos2 += v2; cnt++; }
        xp += bstride;
    }

    const float pc = fmaxf((float)cnt, 1.f);
    const float ac = fmaxf((float)(B_SZ - cnt), 1.f);

    float posv[3] = {pos0, pos1, pos2};
    float totv[3] = {tot0, tot1, tot2};
#pragma unroll
    for (int i = 0; i < 3; ++i) {
        const int d = tid + i * 256;
        const float gp = G[(size_t)l * D_SZ + d];
        const float ga = G[(size_t)L_SZ * D_SZ + d];     // last prototype row
        const float ap = 0.5f * (posv[i] / pc) + 0.5f * gp;
        const float an = 0.5f * ((totv[i] - posv[i]) / ac) + 0.5f * ga;
        Amat[(size_t)l * D_SZ + d] = ap;
        Amat[(size_t)(L_SZ + l) * D_SZ + d] = an;
    }
    if (tid == 0) posCnt[l] = (float)cnt;
}

// ---------------------------------------------------------------------------
// Kernels 2/3: fp32 WMMA GEMM:  C[m,n] = sum_k A[m,k] * W[n,k] + bias[n]
// (optionally ReLU). A: [M x K] row-major, W: [N x K] row-major.
// Each wave computes a 32(M) x 64(N) tile: 2 A frags + 4 B frags feed
// 8 x v_wmma_f32_16x16x4_f32 per K-step (8 FLOP/byte at 32x32 -> ~10.7 FLOP/byte here).
// Block = 256 threads = 8 waves arranged 2(M) x 4(N) -> 64 x 256 per block.
// ---------------------------------------------------------------------------
template <int DO_RELU>
__global__ __launch_bounds__(256) void gemm_wmma(
    const float* __restrict__ A,     // [M, K]
    const float* __restrict__ W,     // [N, K]
    const float* __restrict__ bias,  // [N]
    float* __restrict__ C,           // [M, N]
    int M, int N, int K)
{
    const int tid  = threadIdx.x;
    const int lane = tid & 31;
    const int wave = tid >> 5;        // 0..7
    const int wm   = wave & 1;        // 0..1  (M dir)
    const int wn   = wave >> 1;       // 0..3  (N dir)
    const int row  = lane & 15;       // M-row (A) / N-row (B) / N-col (C)
    const int half = lane >> 4;       // selects K pair {0,1} vs {2,3}

    const int m0 = blockIdx.x * 64  + wm * 32;
    const int n0 = blockIdx.y * 256 + wn * 64;

    v8f acc[2][4];
#pragma unroll
    for (int mi = 0; mi < 2; ++mi)
#pragma unroll
        for (int ni = 0; ni < 4; ++ni)
            acc[mi][ni] = (v8f){};

    // Per-lane fragment base pointers. ISA f32 A-layout (16x4):
    //   VGPR0: lanes0-15 K=0, lanes16-31 K=2 ; VGPR1: K=1 / K=3
    // -> each lane reads float2 at column (k0 + 2*half). Same addressing for
    //    the B operand since B[k][n] = W[n][k] with n striped over lanes 0..15.
    const float* ap[2];
    const float* bp[4];
#pragma unroll
    for (int mi = 0; mi < 2; ++mi)
        ap[mi] = A + (size_t)(m0 + 16 * mi + row) * K + 2 * half;
#pragma unroll
    for (int ni = 0; ni < 4; ++ni)
        bp[ni] = W + (size_t)(n0 + 16 * ni + row) * K + 2 * half;

#pragma unroll 2
    for (int k = 0; k < K; k += 4) {
        v2f a[2], b[4];
#pragma unroll
        for (int mi = 0; mi < 2; ++mi) a[mi] = *(const v2f*)(ap[mi] + k);
#pragma unroll
        for (int ni = 0; ni < 4; ++ni) b[ni] = *(const v2f*)(bp[ni] + k);
#pragma unroll
        for (int mi = 0; mi < 2; ++mi)
#pragma unroll
            for (int ni = 0; ni < 4; ++ni)
                acc[mi][ni] = __builtin_amdgcn_wmma_f32_16x16x4_f32(
                    false, a[mi], false, b[ni], (short)0, acc[mi][ni],
                    false, false);
    }

    // C layout: VGPR r -> lanes0-15: M=r, N=lane ; lanes16-31: M=8+r, N=lane-16
#pragma unroll
    for (int mi = 0; mi < 2; ++mi) {
#pragma unroll
        for (int ni = 0; ni < 4; ++ni) {
            const int n  = n0 + 16 * ni + row;
            const int mb = m0 + 16 * mi + 8 * half;
            const float bv = bias[n];
#pragma unroll
            for (int r = 0; r < 8; ++r) {
                float v = acc[mi][ni][r] + bv;
                if (DO_RELU) v = v > 0.f ? v : 0.f;
                C[(size_t)(mb + r) * N + n] = v;
            }
        }
    }
}

// ---------------------------------------------------------------------------
// Kernel 4: zero proto rows with no positive members; copy rest to output.
// ---------------------------------------------------------------------------
__global__ __launch_bounds__(256) void proto_mask(
    const float* __restrict__ P,       // [2048, 768]
    const float* __restrict__ posCnt,  // [1024]
    float* __restrict__ out)           // [1025, 768]
{
    const int l   = blockIdx.x;
    const int tid = threadIdx.x;
    const float keep = (posCnt[l] > 0.f) ? 1.f : 0.f;
#pragma unroll
    for (int i = 0; i < 3; ++i) {
        const int d = tid + i * 256;
        out[(size_t)l * D_SZ + d] = keep * P[(size_t)l * D_SZ + d];
    }
}

// ---------------------------------------------------------------------------
// Kernel 5: weighted mean of anti-proto rows (weight = anti_cnt > 0).
// grid 6 x 128 threads: one thread per output column.
// ---------------------------------------------------------------------------
__global__ __launch_bounds__(128) void anti_reduce(
    const float* __restrict__ P,       // [2048, 768]
    const float* __restrict__ posCnt,  // [1024]
    float* __restrict__ out)           // [1025, 768]
{
    const int d = blockIdx.x * 128 + threadIdx.x;
    float s = 0.f, ws = 0.f;
    for (int l = 0; l < L_SZ; ++l) {
        const float w = (posCnt[l] < (float)B_SZ) ? 1.f : 0.f;  // anti_cnt > 0
        s  += w * P[(size_t)(L_SZ + l) * D_SZ + d];
        ws += w;
    }
    out[(size_t)L_SZ * D_SZ + d] = s / fmaxf(ws, 1.f);
}

extern "C" void kernel_launch(void* const* d_in, const int* in_sizes, int n_in,
                              void* d_out, int out_size, void* d_ws, size_t ws_size,
                              hipStream_t stream) {
    (void)in_sizes; (void)n_in; (void)out_size; (void)ws_size;
    const float* x    = (const float*)d_in[0];
    const int*   mask = (const int*)  d_in[1];
    const float* G    = (const float*)d_in[2];
    const float* W_h  = (const float*)d_in[3];
    const float* b_h  = (const float*)d_in[4];
    const float* W_p  = (const float*)d_in[5];
    const float* b_p  = (const float*)d_in[6];
    float* out = (float*)d_out;

    float* Amat   = (float*)d_ws;                       // [2048 x 768]
    float* Hmat   = Amat + (size_t)M_ROWS * D_SZ;       // [2048 x 768]
    float* Pmat   = Hmat + (size_t)M_ROWS * H_SZ;       // [2048 x 768]
    float* posCnt = Pmat + (size_t)M_ROWS * D_SZ;       // [1024]

    // Stage 1: masked reduction over B + prototype averaging (memory-bound).
    reduce_kernel<<<L_SZ, 256, 0, stream>>>(x, mask, G, Amat, posCnt);

    // Stage 2: MLP via fp32 WMMA GEMMs (block tile 64x256, wave tile 32x64).
    dim3 gemm_grid(M_ROWS / 64, D_SZ / 256);
    gemm_wmma<1><<<gemm_grid, 256, 0, stream>>>(Amat, W_h, b_h, Hmat,
                                                M_ROWS, H_SZ, D_SZ);
    gemm_wmma<0><<<gemm_grid, 256, 0, stream>>>(Hmat, W_p, b_p, Pmat,
                                                M_ROWS, D_SZ, H_SZ);

    // Stage 3: finalize outputs.
    proto_mask<<<L_SZ, 256, 0, stream>>>(Pmat, posCnt, out);
    anti_reduce<<<D_SZ / 128, 128, 0, stream>>>(Pmat, posCnt, out);
}